// BoundaryLoss_28819230556454
// MI455X (gfx1250) — compile-verified
//
#include <hip/hip_runtime.h>

#define NN 112
#define SS2 (112 * 112)
#define SS3 (112 * 112 * 112)
#define NBC 8                     // B*C = 2*4
#define NB 2
#define NV (NBC * SS3)            // 11,239,424 voxels
#define BIGF 1e10f

typedef float v2f __attribute__((ext_vector_type(2)));
typedef float v8f __attribute__((ext_vector_type(8)));

// ---------------------------------------------------------------------------
// Init: fA = dist-field seeded at background (0 where gt==0, BIG where gt!=0)
//       fB = dist-field seeded at foreground. Also per-(b,c) hasPos flag.
// Each block covers 256 contiguous voxels; SS3 % 256 == 0 so bc is uniform.
// ---------------------------------------------------------------------------
__global__ void init_fields(const int* __restrict__ gt,
                            float* __restrict__ fA,
                            float* __restrict__ fB,
                            float* __restrict__ hasPos) {
  __shared__ int anyp;
  const int tid = threadIdx.x;
  if (tid == 0) anyp = 0;
  __syncthreads();
  const int base = blockIdx.x * 256;
  const int i = base + tid;
  const bool pos = (gt[i] != 0);
  fA[i] = pos ? BIGF : 0.0f;
  fB[i] = pos ? 0.0f : BIGF;
  if (pos) anyp = 1;            // idempotent racy LDS store
  __syncthreads();
  if (tid == 0 && anyp) hasPos[base / SS3] = 1.0f;  // idempotent global store
}

// ---------------------------------------------------------------------------
// Pass along W (contiguous axis). One block = 16 consecutive lines (1792
// contiguous floats), processed in-place via an LDS tile for both fields.
// out[i] = min_j f[j] + (i-j)^2
// ---------------------------------------------------------------------------
__global__ void pass_w(float* __restrict__ fA, float* __restrict__ fB) {
  __shared__ float tile[16 * NN];
  const int tid = threadIdx.x;
  const int t = blockIdx.x;
  const int bc  = t / (112 * 7);
  const int rem = t % (112 * 7);
  const int d   = rem / 7;
  const int g   = rem % 7;
  const int base = bc * SS3 + d * SS2 + g * (16 * NN);

  float* bufs[2] = {fA, fB};
  for (int f = 0; f < 2; ++f) {
    float* p = bufs[f];
    for (int e = tid; e < 16 * NN; e += 256) tile[e] = p[base + e];
    __syncthreads();

    const int L  = tid >> 4;         // line within tile (0..15)
    const int i0 = tid & 15;         // first output index (0..15)
    const float* line = &tile[L * NN];
    float acc[7];
#pragma unroll
    for (int k = 0; k < 7; ++k) acc[k] = BIGF;

#pragma unroll 4
    for (int j = 0; j < NN; ++j) {
      const float fj = line[j];      // LDS broadcast (2 addrs per wave)
#pragma unroll
      for (int k = 0; k < 7; ++k) {
        const float dd = (float)(i0 + 16 * k - j);
        acc[k] = fminf(acc[k], fj + dd * dd);
      }
    }
#pragma unroll
    for (int k = 0; k < 7; ++k)
      p[base + L * NN + i0 + 16 * k] = acc[k];   // coalesced 64B segments
    __syncthreads();
  }
}

// ---------------------------------------------------------------------------
// Pass along a strided axis (H: axisStride=112, outerStride=112^2;
//                            D: axisStride=112^2, outerStride=112).
// One block = 16-wide stripe of lines x full 112 axis, in-place via LDS tile.
// ---------------------------------------------------------------------------
__global__ void pass_s(float* __restrict__ fA, float* __restrict__ fB,
                       int axisStride, int outerStride) {
  __shared__ float tile[NN * 16];    // [j][w]
  const int tid = threadIdx.x;
  const int t = blockIdx.x;
  const int bc  = t / (112 * 7);
  const int rem = t % (112 * 7);
  const int o2  = rem / 7;
  const int tw  = rem % 7;
  const int base = bc * SS3 + o2 * outerStride + tw * 16;

  float* bufs[2] = {fA, fB};
  for (int f = 0; f < 2; ++f) {
    float* p = bufs[f];
    for (int e = tid; e < NN * 16; e += 256) {
      const int j = e >> 4, w = e & 15;
      tile[e] = p[base + j * axisStride + w];
    }
    __syncthreads();

    const int w  = tid & 15;
    const int i0 = tid >> 4;
    float acc[7];
#pragma unroll
    for (int k = 0; k < 7; ++k) acc[k] = BIGF;

#pragma unroll 4
    for (int j = 0; j < NN; ++j) {
      const float fj = tile[j * 16 + w];   // conflict-free: 16 banks, 2x bcast
#pragma unroll
      for (int k = 0; k < 7; ++k) {
        const float dd = (float)(i0 + 16 * k - j);
        acc[k] = fminf(acc[k], fj + dd * dd);
      }
    }
#pragma unroll
    for (int k = 0; k < 7; ++k)
      p[base + (i0 + 16 * k) * axisStride + w] = acc[k];
    __syncthreads();
  }
}

// ---------------------------------------------------------------------------
// Reduce: softmax over channels, contrib = sum_c p_c * |sdf_c| (gated by
// hasPos), then wave-level reduction through V_WMMA_F32_16X16X4_F32:
// A(16x4) holds lane partials, B = ones -> C rows hold row-sums; summing the
// 8 C VGPRs per lane + one shfl_xor(16) yields the full 32-lane sum in f32.
// ---------------------------------------------------------------------------
__global__ void reduce_k(const float* __restrict__ logits,
                         const float* __restrict__ fA,
                         const float* __restrict__ fB,
                         const float* __restrict__ hasPos,
                         float* __restrict__ accOut) {
  const int item = blockIdx.x * 256 + threadIdx.x;   // item < NB*SS3 (exact)
  const int b = item / SS3;
  const int x = item - b * SS3;

  const float* lb = logits + (b * 4) * SS3 + x;
  float l[4];
#pragma unroll
  for (int c = 0; c < 4; ++c) l[c] = lb[c * SS3];
  float m = fmaxf(fmaxf(l[0], l[1]), fmaxf(l[2], l[3]));
  float e[4], s = 0.0f;
#pragma unroll
  for (int c = 0; c < 4; ++c) { e[c] = __expf(l[c] - m); s += e[c]; }
  const float inv = 1.0f / s;

  float contrib = 0.0f;
#pragma unroll
  for (int c = 0; c < 4; ++c) {
    const int idx = (b * 4 + c) * SS3 + x;
    const float sdf = sqrtf(fB[idx]) - sqrtf(fA[idx]);  // d_out - d_in
    contrib += (e[c] * inv) * fabsf(sdf) * hasPos[b * 4 + c];
  }

  // --- matrix-engine wave reduction (all 32 lanes active by construction) ---
  v2f a;  a.x = contrib; a.y = 0.0f;      // A: row (lane%16), K striped
  v2f bb; bb.x = 1.0f;  bb.y = 1.0f;      // B: all ones
  v8f cc = {};
  cc = __builtin_amdgcn_wmma_f32_16x16x4_f32(
      /*neg_a=*/false, a, /*neg_b=*/false, bb,
      /*c_mod=*/(short)0, cc, /*reuse_a=*/false, /*reuse_b=*/false);
  float sl = cc[0] + cc[1] + cc[2] + cc[3] + cc[4] + cc[5] + cc[6] + cc[7];
  sl += __shfl_xor(sl, 16);               // rows 0-7 half + rows 8-15 half

  __shared__ float wsum[8];
  const int lane = threadIdx.x & 31, wv = threadIdx.x >> 5;
  if (lane == 0) wsum[wv] = sl;
  __syncthreads();
  if (threadIdx.x == 0) {
    float bsum = 0.0f;
#pragma unroll
    for (int k = 0; k < 8; ++k) bsum += wsum[k];
    atomicAdd(accOut, bsum);
  }
}

__global__ void finalize_k(const float* __restrict__ acc, float* __restrict__ out) {
  out[0] = acc[0] * (1.0f / (float)NV);
}

// ---------------------------------------------------------------------------
extern "C" void kernel_launch(void* const* d_in, const int* in_sizes, int n_in,
                              void* d_out, int out_size, void* d_ws, size_t ws_size,
                              hipStream_t stream) {
  const float* logits = (const float*)d_in[0];
  const int*   gt     = (const int*)d_in[1];

  float* acc    = (float*)d_ws;               // [0]    running sum
  float* hasPos = (float*)d_ws + 4;           // [4..11] per-(b,c) flags
  float* fA = (float*)((char*)d_ws + 256);    // dist^2 to nearest background
  float* fB = fA + NV;                        // dist^2 to nearest foreground

  hipMemsetAsync(d_ws, 0, 256, stream);

  init_fields<<<NV / 256, 256, 0, stream>>>(gt, fA, fB, hasPos);

  const int passBlocks = NBC * 112 * 7;       // 6272
  pass_w<<<passBlocks, 256, 0, stream>>>(fA, fB);                 // W axis
  pass_s<<<passBlocks, 256, 0, stream>>>(fA, fB, NN, SS2);        // H axis
  pass_s<<<passBlocks, 256, 0, stream>>>(fA, fB, SS2, NN);        // D axis

  reduce_k<<<(NB * SS3) / 256, 256, 0, stream>>>(logits, fA, fB, hasPos, acc);
  finalize_k<<<1, 1, 0, stream>>>(acc, (float*)d_out);
}